// GramMatrix_412316860971
// MI455X (gfx1250) — compile-verified
//
#include <hip/hip_runtime.h>
#include <hip/hip_bf16.h>
#include <stdint.h>

// GramMatrix: out[i,j] = sigmoid( sum_d W[d]*|x[i,d]-x[j,d]| + b ), N=2048, D=64.
// L1 kernel => |x_i - x_j| is not A[i,k]*B[k,j]-factorable => the core runs on
// VALU (v_sub + v_fma with |src| modifier, 2 ops/point = the f32 floor).
// CDNA5-specific paths used:
//   * Tensor Data Mover: tensor_load_to_lds stages x-tiles into LDS, with the
//     TDM pad feature (4 dwords every 64) writing rows at stride 68 so all
//     ds_load_b128 reads are bank-conflict-free. Completion via s_wait_tensorcnt.
//   * v_wmma_f32_16x16x4_f32: exact bias-add epilogue on the matrix pipe
//     (A = bias/4, B = 1.0  =>  D = bias + C, bit-exact), with accumulators
//     laid out in the native 16x16 f32 C/D VGPR layout (VGPR v <-> rows v,v+8).
//   * Fast sigmoid epilogue: v_exp_f32 + v_rcp_f32 (no IEEE division sequence).

typedef unsigned int u32;
typedef __attribute__((ext_vector_type(4))) u32 sgpr4_t;  // -> SGPR quad  s[n:n+3]
typedef __attribute__((ext_vector_type(8))) u32 sgpr8_t;  // -> SGPR octet s[n:n+7]
typedef __attribute__((ext_vector_type(2))) float v2f;
typedef __attribute__((ext_vector_type(8))) float v8f;

#define D_DIM       64
#define TILE_I      128
#define TILE_J      64
#define LDS_STRIDE  68   // 64 data dwords + 4 pad dwords inserted by the TDM

// 1D contiguous TDM copy of n_elems f32 from global to LDS, inserting a 4-dword
// pad after every 64 dwords (=> row stride 68 floats in LDS).
__device__ __forceinline__ void tdm_load_to_lds_1d(u32 lds_byte_off,
                                                   const float* __restrict__ gsrc,
                                                   u32 n_elems) {
  const uint64_t ga = (uint64_t)(uintptr_t)gsrc;
  sgpr4_t g0;
  g0[0] = 1u;                                    // count=1 (valid), user mode
  g0[1] = lds_byte_off;                          // lds_addr (bytes)
  g0[2] = (u32)ga;                               // global_addr[31:0]
  g0[3] = (u32)((ga >> 32) & 0x01FFFFFFu)        // global_addr[56:32]
        | (2u << 30);                            // type = 2 ("image")
  sgpr8_t g1;
  g1[0] = (2u << 16)                             // data_size: 4 bytes
        | (1u << 20)                             // pad_enable
        | (5u << 22)                             // pad_interval code 5 = 64 dwords
        | (3u << 25);                            // pad_amount  code 3 = 4 dwords
  g1[1] = (n_elems & 0xFFFFu) << 16;             // tensor_dim0[15:0]
  g1[2] = (n_elems >> 16) | (1u << 16);          // tensor_dim0[31:16] | tensor_dim1=1
  g1[3] = (n_elems & 0xFFFFu) << 16;             // tensor_dim1 hi=0 | tile_dim0
  g1[4] = 1u;                                    // tile_dim1=1 | tile_dim2=0
  g1[5] = n_elems;                               // tensor_dim0_stride[31:0]
  g1[6] = 0u;
  g1[7] = 0u;
  asm volatile("tensor_load_to_lds %0, %1" :: "s"(g0), "s"(g1) : "memory");
}

__device__ __forceinline__ void wait_tensorcnt0() {
#if __has_builtin(__builtin_amdgcn_s_wait_tensorcnt)
  __builtin_amdgcn_s_wait_tensorcnt(0);
#else
  asm volatile("s_wait_tensorcnt 0x0" ::: "memory");
#endif
}

__global__ __launch_bounds__(256) void
GramMatrix_l1_sigmoid_kernel(const float* __restrict__ x,
                             const float* __restrict__ W,
                             const float* __restrict__ b,
                             float* __restrict__ out, int n) {
  __shared__ __align__(16) float shW[D_DIM];
  __shared__ __align__(16) float shI[TILE_I * LDS_STRIDE];
  __shared__ __align__(16) float shJ[TILE_J * LDS_STRIDE];

  const int t    = threadIdx.x;
  const int lane = t & 31;        // wave32 lane
  const int wv   = t >> 5;        // wave id 0..7
  const int tx   = lane & 15;     // WMMA N index: j = jBase + tx + 16*c
  const int half = lane >> 4;     // WMMA row-half: i = iBase + 16*wv + 8*half + v
  const int iBase = blockIdx.y * TILE_I;
  const int jBase = blockIdx.x * TILE_J;
  const int rI = (wv << 4) + (half << 3);   // this lane's i-row base within shI

  if (t < D_DIM) shW[t] = W[t];

  if (t < 32) {  // one wave drives the Tensor Data Mover
    tdm_load_to_lds_1d((u32)(uintptr_t)&shI[0],
                       x + (size_t)iBase * D_DIM, TILE_I * D_DIM);
    tdm_load_to_lds_1d((u32)(uintptr_t)&shJ[0],
                       x + (size_t)jBase * D_DIM, TILE_J * D_DIM);
    wait_tensorcnt0();
  }
  __syncthreads();

  // acc[v][c] sits exactly in WMMA 16x16 f32 C/D layout per c:
  //   VGPR v, lanes 0-15  -> M = v      (i = iBase+16*wv+v),     N = lane
  //   VGPR v, lanes 16-31 -> M = v + 8  (i = iBase+16*wv+8+v),   N = lane-16
  float acc[8][4];
#pragma unroll
  for (int v = 0; v < 8; ++v)
#pragma unroll
    for (int c = 0; c < 4; ++c) acc[v][c] = 0.0f;

  for (int d4 = 0; d4 < D_DIM / 4; ++d4) {
    const float4 w4 = *(const float4*)&shW[d4 * 4];
    float4 aj[4];
#pragma unroll
    for (int c = 0; c < 4; ++c)
      aj[c] = *(const float4*)&shJ[(tx + 16 * c) * LDS_STRIDE + d4 * 4];
#pragma unroll
    for (int v = 0; v < 8; ++v) {
      const float4 ai = *(const float4*)&shI[(rI + v) * LDS_STRIDE + d4 * 4];
#pragma unroll
      for (int c = 0; c < 4; ++c) {
        acc[v][c] = fmaf(w4.x, __builtin_fabsf(ai.x - aj[c].x), acc[v][c]);
        acc[v][c] = fmaf(w4.y, __builtin_fabsf(ai.y - aj[c].y), acc[v][c]);
        acc[v][c] = fmaf(w4.z, __builtin_fabsf(ai.z - aj[c].z), acc[v][c]);
        acc[v][c] = fmaf(w4.w, __builtin_fabsf(ai.w - aj[c].w), acc[v][c]);
      }
    }
  }

  const float bias = b[0];
#pragma unroll
  for (int c = 0; c < 4; ++c) {
    v8f c8;
#pragma unroll
    for (int v = 0; v < 8; ++v) c8[v] = acc[v][c];
#if __has_builtin(__builtin_amdgcn_wmma_f32_16x16x4_f32)
    // Exact bias add on the matrix pipe: sum_k (bias/4)*1.0 == bias (RNE-exact,
    // power-of-two scaling), so D = bias + C bit-exactly. EXEC is all-1s here.
    v2f a2; a2[0] = bias * 0.25f; a2[1] = bias * 0.25f;   // A: 16x4, all bias/4
    v2f b2; b2[0] = 1.0f;        b2[1] = 1.0f;            // B: 4x16, all 1.0
    c8 = __builtin_amdgcn_wmma_f32_16x16x4_f32(false, a2, false, b2,
                                               (short)0, c8, false, false);
#else
#pragma unroll
    for (int v = 0; v < 8; ++v) c8[v] += bias;
#endif
#pragma unroll
    for (int v = 0; v < 8; ++v) {
      // sigmoid via v_exp_f32 + v_rcp_f32 (avoid the IEEE div_scale/div_fixup
      // expansion; ~1 ulp, plenty for a sigmoid output).
      const float e   = __expf(-c8[v]);
      const float sig = __builtin_amdgcn_rcpf(1.0f + e);
      const size_t row = (size_t)(iBase + rI + v) * (size_t)n;
      __builtin_nontemporal_store(sig, &out[row + jBase + tx + 16 * c]);
    }
  }
}

extern "C" void kernel_launch(void* const* d_in, const int* in_sizes, int n_in,
                              void* d_out, int out_size, void* d_ws, size_t ws_size,
                              hipStream_t stream) {
  const float* x = (const float*)d_in[0];  // [N, 64] f32
  const float* W = (const float*)d_in[1];  // [64, 1] f32
  const float* b = (const float*)d_in[2];  // [1]     f32
  float* out = (float*)d_out;              // [N, N]  f32

  const int n = in_sizes[0] / D_DIM;       // N = 2048
  dim3 grid(n / TILE_J, n / TILE_I);       // 32 x 16 = 512 workgroups
  GramMatrix_l1_sigmoid_kernel<<<grid, 256, 0, stream>>>(x, W, b, out, n);
}